// SSIMLoss_17325898072140
// MI455X (gfx1250) — compile-verified
//
#include <hip/hip_runtime.h>

typedef __attribute__((ext_vector_type(2))) float v2f;
typedef __attribute__((ext_vector_type(8))) float v8f;

#define H 512
#define W 512
#define OUTD 502          // 512 - 11 + 1
#define TILE_R 32         // output rows per block
#define TILE_C 48         // output cols per block
#define XR 44             // input rows staged (32 + 10, padded to K=28 reach)
#define XP 64             // input col pitch (48 + 10 -> padded to 4 N-chunks)
#define TP 66             // T pitch (64 cols produced, +2 pad: kills bank conflicts)
#define NCHUNK 7          // K = 28 = 7 x 4 (covers 11-tap reach of 26)

// 11-tap gaussian, sigma=1.5, normalized (matches pytorch-msssim defaults)
__device__ __constant__ float G11[11] = {
    0.00102842f, 0.00759877f, 0.03600078f, 0.10936060f, 0.21300560f,
    0.26601171f,
    0.21300560f, 0.10936060f, 0.03600078f, 0.00759877f, 0.00102842f};

__device__ __forceinline__ v8f wmma_f32(v2f a, v2f b, v8f c) {
  // D = A(16x4,f32) * B(4x16,f32) + C(16x16,f32)
  return __builtin_amdgcn_wmma_f32_16x16x4_f32(
      false, a, false, b, (short)0, c, false, false);
}

extern "C" __global__ __launch_bounds__(256) void ssim_tile_kernel(
    const float* __restrict__ pred, const float* __restrict__ tgt,
    float* __restrict__ imgsum) {
  __shared__ float sX[XR * XP];
  __shared__ float sY[XR * XP];
  __shared__ float sTX[TILE_R * TP];
  __shared__ float sTY[TILE_R * TP];
  __shared__ float sTXX[TILE_R * TP];
  __shared__ float sTYY[TILE_R * TP];
  __shared__ float sTXY[TILE_R * TP];
  __shared__ float sG[16];
  __shared__ float sSum;

  const int tid = threadIdx.x;
  const int img = blockIdx.z;
  const int r0 = blockIdx.y * TILE_R;
  const int c0 = blockIdx.x * TILE_C;
  const size_t ib = (size_t)img * (H * W);

  if (tid < 16) sG[tid] = (tid < 11) ? G11[tid] : 0.0f;
  if (tid == 0) sSum = 0.0f;

  // ---- stage 0: stage inputs to LDS with fused unnormalize + nan_to_num ----
  for (int idx = tid; idx < XR * XP; idx += 256) {
    const int r = idx / XP, c = idx % XP;
    const int gr = r0 + r, gc = c0 + c;
    float x = 0.0f, y = 0.0f;
    if (gr < H && gc < W) {
      x = pred[ib + (size_t)gr * W + gc];
      y = tgt[ib + (size_t)gr * W + gc];
    }
    x = x * 8.0f + 13.5f;  x = (x + 2.0f) * (1.0f / 37.0f);
    y = y * 8.0f + 13.5f;  y = (y + 2.0f) * (1.0f / 37.0f);
    x = (x == x) ? x : 0.0f;   // nan_to_num
    y = (y == y) ? y : 0.0f;
    sX[idx] = x;
    sY[idx] = y;
  }
  __syncthreads();

  const int lane = tid & 31;
  const int wid = tid >> 5;
  const int m = lane & 15;               // M (vertical-A) / N (horizontal-B)
  const int khalf = (lane >> 4) << 1;    // K split {0,1}/{2,3} across halves
  const int rhalf = (lane >> 4) << 3;    // D-layout row split {0..7}/{8..15}

  // Banded gaussian operand, identical for vertical-A and horizontal-B:
  // Wk[c] supplies W[m][k] = g[k - m], k = c*4 + khalf + {0,1}
  v2f Wk[NCHUNK];
#pragma unroll
  for (int c = 0; c < NCHUNK; ++c) {
    const int k0 = c * 4 + khalf;
    const int d0 = k0 - m, d1 = d0 + 1;
    float w0 = ((unsigned)d0 < 11u) ? sG[d0 & 15] : 0.0f;
    float w1 = ((unsigned)d1 < 11u) ? sG[d1 & 15] : 0.0f;
    v2f w; w.x = w0; w.y = w1;
    Wk[c] = w;
  }

  // ---- stage 1: vertical blur via WMMA (8 tiles, 1 per wave) ----
  // T[16x16] = Wv(16x28) x In(28x16); maps X,Y,XX,YY,XY share the loads.
  {
    const int t = wid;                   // 2 row-tiles x 4 col-chunks
    const int rb = (t >> 2) * 16;
    const int cb = (t & 3) * 16;
    v8f aX = {}, aY = {}, aXX = {}, aYY = {}, aXY = {};
#pragma unroll
    for (int c = 0; c < NCHUNK; ++c) {
      const int kr = rb + c * 4 + khalf;
      const float x0 = sX[kr * XP + cb + m];
      const float x1 = sX[(kr + 1) * XP + cb + m];
      const float y0 = sY[kr * XP + cb + m];
      const float y1 = sY[(kr + 1) * XP + cb + m];
      v2f bx, by, bxx, byy, bxy;
      bx.x = x0;       bx.y = x1;
      by.x = y0;       by.y = y1;
      bxx.x = x0 * x0; bxx.y = x1 * x1;
      byy.x = y0 * y0; byy.y = y1 * y1;
      bxy.x = x0 * y0; bxy.y = x1 * y1;
      aX  = wmma_f32(Wk[c], bx,  aX);
      aY  = wmma_f32(Wk[c], by,  aY);
      aXX = wmma_f32(Wk[c], bxx, aXX);
      aYY = wmma_f32(Wk[c], byy, aYY);
      aXY = wmma_f32(Wk[c], bxy, aXY);
    }
#pragma unroll
    for (int i = 0; i < 8; ++i) {
      const int off = (rb + i + rhalf) * TP + cb + m;
      sTX[off] = aX[i];
      sTY[off] = aY[i];
      sTXX[off] = aXX[i];
      sTYY[off] = aYY[i];
      sTXY[off] = aXY[i];
    }
  }
  __syncthreads();

  // ---- stage 2: horizontal blur via WMMA + fused SSIM (6 tiles) ----
  if (wid < 6) {
    const int t = wid;                   // row-tile t>>2 x col-tile t&3 (4 + 2 tiles)
    const int rb = (t >> 2) * 16;        // t in 0..5 -> rb 0 or 16
    const int cb = (t & 3) * 16;
    v8f aX = {}, aY = {}, aXX = {}, aYY = {}, aXY = {};
#pragma unroll
    for (int c = 0; c < NCHUNK; ++c) {
      const int off = (rb + m) * TP + cb + c * 4 + khalf; // 8B aligned
      const v2f tX  = *(const v2f*)&sTX[off];
      const v2f tY  = *(const v2f*)&sTY[off];
      const v2f tXX = *(const v2f*)&sTXX[off];
      const v2f tYY = *(const v2f*)&sTYY[off];
      const v2f tXY = *(const v2f*)&sTXY[off];
      aX  = wmma_f32(tX,  Wk[c], aX);
      aY  = wmma_f32(tY,  Wk[c], aY);
      aXX = wmma_f32(tXX, Wk[c], aXX);
      aYY = wmma_f32(tYY, Wk[c], aYY);
      aXY = wmma_f32(tXY, Wk[c], aXY);
    }
    const float C1v = 0.0001f;  // (0.01*1)^2
    const float C2v = 0.0009f;  // (0.03*1)^2
    float s = 0.0f;
#pragma unroll
    for (int i = 0; i < 8; ++i) {
      const float mu1 = aX[i], mu2 = aY[i];
      const float m11 = mu1 * mu1, m22 = mu2 * mu2, m12 = mu1 * mu2;
      const float s1 = aXX[i] - m11;
      const float s2 = aYY[i] - m22;
      const float s12 = aXY[i] - m12;
      const float v = ((2.0f * m12 + C1v) / (m11 + m22 + C1v)) *
                      ((2.0f * s12 + C2v) / (s1 + s2 + C2v));
      const int gr = r0 + rb + i + rhalf;
      const int gc = c0 + cb + m;
      s += (gr < OUTD && gc < OUTD) ? v : 0.0f;
    }
    atomicAdd(&sSum, s);                 // ds_add_f32
  }
  __syncthreads();
  if (tid == 0) atomicAdd(&imgsum[img], sSum);  // global_atomic_add_f32
}

extern "C" __global__ __launch_bounds__(64) void ssim_final_kernel(
    const float* __restrict__ imgsum, float* __restrict__ out) {
  __shared__ float red[64];
  const int t = threadIdx.x;
  float v = imgsum[t] * (1.0f / ((float)OUTD * (float)OUTD));
  v = fmaxf(v, 0.0f);                    // nonnegative_ssim
  red[t] = v;
  __syncthreads();
  for (int s = 32; s > 0; s >>= 1) {
    if (t < s) red[t] += red[t + s];
    __syncthreads();
  }
  if (t == 0) out[0] = red[0] * (1.0f / 64.0f);
}

extern "C" void kernel_launch(void* const* d_in, const int* in_sizes, int n_in,
                              void* d_out, int out_size, void* d_ws,
                              size_t ws_size, hipStream_t stream) {
  (void)in_sizes; (void)n_in; (void)out_size; (void)ws_size;
  const float* pred = (const float*)d_in[0];
  const float* tgt = (const float*)d_in[1];
  float* imgsum = (float*)d_ws;  // 64 per-image partial sums
  hipMemsetAsync(imgsum, 0, 64 * sizeof(float), stream);
  dim3 grid((OUTD + TILE_C - 1) / TILE_C,   // 11
            (OUTD + TILE_R - 1) / TILE_R,   // 16
            64);
  ssim_tile_kernel<<<grid, 256, 0, stream>>>(pred, tgt, imgsum);
  ssim_final_kernel<<<1, 64, 0, stream>>>(imgsum, (float*)d_out);
}